// GaussianAttention_31971736552125
// MI455X (gfx1250) — compile-verified
//
#include <hip/hip_runtime.h>
#include <hip/hip_bf16.h>

// out[b,s,i,j] = exp( -(sum_d |x[b,i,d]-x[b,j,d]|)^2 / (2*sigma_s^2) )
// B=2, N=2048, D=3, S=8 -> 256MB f32 output: store-bandwidth bound (~11us @ 23.3TB/s).
// One block per (b,i) row. TDM stages x[b] (24KB) into LDS once per block.

#define DIM      3
#define MAXN     2048
#define NSIG_MAX 16
#define THREADS  256
#define USE_TDM  1

typedef unsigned int       u32;
typedef unsigned long long u64;
typedef u32   sgpr4 __attribute__((ext_vector_type(4)));
typedef u32   sgpr8 __attribute__((ext_vector_type(8)));
typedef float v4f   __attribute__((ext_vector_type(4)));   // native vector for nontemporal b128

template <int S_CT>
__global__ __launch_bounds__(THREADS) void gauss_bank_kernel(
    const float* __restrict__ x,      // [B, N, 3]
    const float* __restrict__ sigmas, // [S]
    float* __restrict__ out,          // [B, S, N, N]
    int nShift,                       // log2(N)
    int S_rt)                         // runtime S (used only when S_CT == 0)
{
  __shared__ float xs[MAXN * DIM];   // 24 KB: x[b,:,:]
  __shared__ float inv_s[NSIG_MAX];  // 1/(2*sigma^2)

  const int S = (S_CT > 0) ? S_CT : S_rt;
  const int N = 1 << nShift;
  const int b = (int)blockIdx.x >> nShift;       // power-of-two N: no SALU division
  const int i = (int)blockIdx.x & (N - 1);
  const float* __restrict__ xb = x + (size_t)b * N * DIM;
  const u32 nelem = (u32)(N * DIM);  // 6144 floats = 24576 bytes

#if USE_TDM
  // ---- Tensor Data Mover: DMA x[b] (1 x nelem tile of 4B elems) into LDS ----
  if (threadIdx.x < 32) {  // one wave issues the descriptor (TDM ignores EXEC)
    const u32 lds_off = (u32)(u64)(uintptr_t)&xs[0];  // low 32b of generic ptr = LDS byte offset
    const u64 ga      = (u64)(uintptr_t)xb;
    sgpr4 g0;
    g0.x = 1u;                                            // count=1 (valid), no gather
    g0.y = lds_off;                                       // lds_addr
    g0.z = (u32)(ga & 0xFFFFFFFFull);                     // global_addr[31:0]
    g0.w = (u32)((ga >> 32) & 0x01FFFFFFull)              // global_addr[56:32]
         | (2u << 30);                                    // type = 2 ("image")
    sgpr8 g1;
    g1.s0 = (2u << 16);                                   // wg_mask=0, data_size=2 (4B)
    g1.s1 = (nelem & 0xFFFFu) << 16;                      // tensor_dim0[15:0]  (bits 63:48)
    g1.s2 = ((nelem >> 16) & 0xFFFFu)                     // tensor_dim0[31:16] (bits 79:64)
          | (1u << 16);                                   // tensor_dim1[15:0] = 1
    g1.s3 = (nelem & 0xFFFFu) << 16;                      // tensor_dim1 hi=0; tile_dim0 = nelem
    g1.s4 = 1u;                                           // tile_dim1 = 1; tile_dim2 = 0
    g1.s5 = nelem;                                        // tensor_dim0_stride[31:0]
    g1.s6 = (nelem & 0xFFFFu) << 16;                      // t0_stride[47:32]=0; t1_stride[15:0]
    g1.s7 = (nelem >> 16) & 0xFFFFu;                      // t1_stride[47:16]
    asm volatile("tensor_load_to_lds %0, %1" :: "s"(g0), "s"(g1) : "memory");
    __builtin_amdgcn_s_wait_tensorcnt(0);                 // issuing wave drains TDM
  }
#else
  for (int t = threadIdx.x; t < N * DIM; t += THREADS) xs[t] = xb[t];
#endif

  if ((int)threadIdx.x < S) {
    const float sg = sigmas[threadIdx.x];
    inv_s[threadIdx.x] = 1.0f / (2.0f * sg * sg);
  }
  __syncthreads();

  const float xi0 = xs[i * DIM + 0];
  const float xi1 = xs[i * DIM + 1];
  const float xi2 = xs[i * DIM + 2];

  const size_t plane = (size_t)N * N;                        // stride between sigma planes
  float* __restrict__ row0 = out + (((size_t)b * S) * N + i) * N;  // s=0 row start

  for (int jj = threadIdx.x; jj < (N >> 2); jj += THREADS) {
    const int j0 = jj << 2;
    float d2[4];
#pragma unroll
    for (int u = 0; u < 4; ++u) {
      const int j = j0 + u;
      const float d = fabsf(xi0 - xs[j * DIM + 0])
                    + fabsf(xi1 - xs[j * DIM + 1])
                    + fabsf(xi2 - xs[j * DIM + 2]);
      d2[u] = d * d;
    }
#pragma unroll
    for (int s = 0; s < ((S_CT > 0) ? S_CT : 1); ++s) {
      // compile-time-unrolled path (S_CT>0); generic path loops below
      if (S_CT > 0) {
        const float c = inv_s[s];
        v4f o;
        o.x = __expf(-d2[0] * c);
        o.y = __expf(-d2[1] * c);
        o.z = __expf(-d2[2] * c);
        o.w = __expf(-d2[3] * c);
        // 256MB write-once stream (> 192MB L2): non-temporal b128 stores
        __builtin_nontemporal_store(o, (v4f*)(row0 + (size_t)s * plane + j0));
      }
    }
    if (S_CT == 0) {
      for (int s = 0; s < S; ++s) {
        const float c = inv_s[s];
        v4f o;
        o.x = __expf(-d2[0] * c);
        o.y = __expf(-d2[1] * c);
        o.z = __expf(-d2[2] * c);
        o.w = __expf(-d2[3] * c);
        __builtin_nontemporal_store(o, (v4f*)(row0 + (size_t)s * plane + j0));
      }
    }
  }
}

extern "C" void kernel_launch(void* const* d_in, const int* in_sizes, int n_in,
                              void* d_out, int out_size, void* d_ws, size_t ws_size,
                              hipStream_t stream) {
  const float* x   = (const float*)d_in[0];   // [B, N, 3] f32
  const float* sig = (const float*)d_in[1];   // [S] f32
  float* out       = (float*)d_out;           // [B, S, N, N] f32

  const int S = in_sizes[1];                  // 8
  // out_size = B*S*N*N ; in_sizes[0] = B*N*3  =>  N = 3*out_size / (S*in_sizes[0])
  const long long N_ll = (3LL * (long long)out_size) / ((long long)S * (long long)in_sizes[0]);
  const int N = (int)N_ll;                    // 2048 (power of two)
  const int B = in_sizes[0] / (N * DIM);      // 2

  int nShift = 0;
  while ((1 << nShift) < N) ++nShift;         // log2(N) = 11

  dim3 grid((unsigned)(B * N));               // one block per (b, i) row
  dim3 block(THREADS);
  if (S == 8) {
    hipLaunchKernelGGL((gauss_bank_kernel<8>), grid, block, 0, stream,
                       x, sig, out, nShift, S);
  } else {
    hipLaunchKernelGGL((gauss_bank_kernel<0>), grid, block, 0, stream,
                       x, sig, out, nShift, S);
  }
}